// DenseNet3D_89730456748628
// MI455X (gfx1250) — compile-verified
//
#include <hip/hip_runtime.h>
#include <math.h>

typedef __attribute__((ext_vector_type(2))) float v2f;
typedef __attribute__((ext_vector_type(8))) float v8f;

__device__ __forceinline__ v8f wmma4(v2f a, v2f b, v8f c) {
  // D = A(16x4 f32) * B(4x16 f32) + C(16x16 f32)
  return __builtin_amdgcn_wmma_f32_16x16x4_f32(false, a, false, b, (short)0, c, false, false);
}

__device__ __forceinline__ v8f v8zero() {
  v8f v;
  #pragma unroll
  for (int i = 0; i < 8; ++i) v[i] = 0.0f;
  return v;
}

__device__ __forceinline__ float sigm(float x) { return 1.0f / (1.0f + expf(-x)); }

// ---------------------------------------------------------------------------
// seq[(t*64+b)*192 + c*4 + r] = x[((b*48+c)*512 + t)*4 + r]
// (materializes transpose(x,(2,0,1,3,4)).reshape(32768,192))
// ---------------------------------------------------------------------------
__global__ void make_seq(const float* __restrict__ x, float* __restrict__ seq) {
  const int total = 32768 * 192;
  for (int idx = blockIdx.x * blockDim.x + threadIdx.x; idx < total;
       idx += gridDim.x * blockDim.x) {
    int f = idx % 192;
    int m = idx / 192;
    int t = m >> 6;
    int b = m & 63;
    int c = f >> 2;
    int r = f & 3;
    seq[idx] = x[(((size_t)(b * 48 + c) * 512) + t) * 4 + r];
  }
}

// ---------------------------------------------------------------------------
// Repack W = [wih | whh] (N=768 x K=KIN+256, row-major per-gate-row) into
// WMMA B-fragment order:  out[((g*48+nt)*32+lane)*2 + j] = Wcat[n][k]
//   n = nt*16 + (lane&15),  k = g*4 + ((lane>>4)<<1) + j
// so the GEMM inner loop does one coalesced b64 load per wave per tile.
// ---------------------------------------------------------------------------
__global__ void reorder_w(const float* __restrict__ wih, const float* __restrict__ whh,
                          float* __restrict__ out, int KIN) {
  const int KTOT = KIN + 256;
  const int total = (KTOT / 4) * 48 * 32 * 2;
  for (int idx = blockIdx.x * blockDim.x + threadIdx.x; idx < total;
       idx += gridDim.x * blockDim.x) {
    int j = idx & 1;
    int l2 = idx >> 1;
    int lane = l2 & 31;
    int t2 = l2 >> 5;
    int nt = t2 % 48;
    int g = t2 / 48;
    int n = nt * 16 + (lane & 15);
    int k = g * 4 + ((lane >> 4) << 1) + j;
    float v = (k < KIN) ? wih[(size_t)n * KIN + k] : whh[(size_t)n * 256 + (k - KIN)];
    out[idx] = v;
  }
}

// ---------------------------------------------------------------------------
// One bidirectional GRU layer, batch-sliced: block = 16 batch rows, 8 waves.
// Waves 0..3: fused r/z gate tiles (x and h contributions summed, 8 tiles ea).
// Waves 4..7: n-gate tiles, x-part and h-part kept separate (4 tiles, 2 accs).
// h state lives in LDS for the whole 64-step scan; no inter-block sync.
// ---------------------------------------------------------------------------
template <int KIN>
__global__ void __launch_bounds__(256)
gru_layer_kernel(const float* __restrict__ xin,                       // (64*512, KIN)
                 const float* __restrict__ wfragF, const float* __restrict__ wfragB,
                 const float* __restrict__ bihF, const float* __restrict__ bhhF,
                 const float* __restrict__ bihB, const float* __restrict__ bhhB,
                 float* __restrict__ yout,                             // (64*512,512) or null
                 float* __restrict__ hfinal) {                         // (512,512) or null
  constexpr int KTOT = KIN + 256;
  constexpr int KG = KTOT / 4;   // k-groups of 4
  constexpr int KGX = KIN / 4;   // k-groups belonging to the x part
  constexpr int HST = 258;       // padded stride for 16x256 buffers
  constexpr int GST = 770;       // padded stride for 16x768 buffer

  __shared__ float Hs[16 * HST];   // hidden state (16 x 256)
  __shared__ float Gh[16 * GST];   // r/z fused preacts (cols 0..511) + h-part of n (512..767)
  __shared__ float Gin[16 * HST];  // x-part of n gate (16 x 256)

  const int tid = threadIdx.x;
  const int lane = tid & 31;
  const int wid = tid >> 5;
  const int dir = blockIdx.y;
  const int n0 = blockIdx.x << 4;

  const float* wfrag = dir ? wfragB : wfragF;
  const float* bih = dir ? bihB : bihF;
  const float* bhh = dir ? bhhB : bhhF;

  for (int i = tid; i < 16 * HST; i += 256) Hs[i] = 0.0f;
  __syncthreads();

  const int mrow = lane & 15;
  const int koff = (lane >> 4) << 1;  // 0 or 2
  const int rbase = (lane >> 4) << 3; // C-frag row base
  const int ncol = lane & 15;

  for (int it = 0; it < 64; ++it) {
    const int s = dir ? (63 - it) : it;
    const float* arow = xin + (size_t)(s * 512 + n0 + mrow) * KIN + koff;
    const float* hrow = Hs + mrow * HST + koff;
    const float2* wf2 = (const float2*)wfrag;

    if (wid < 4) {
      v8f acc[8];
      #pragma unroll
      for (int i = 0; i < 8; ++i) acc[i] = v8zero();
      for (int g = 0; g < KG; ++g) {
        v2f a;
        if (g < KGX) { a[0] = arow[g * 4]; a[1] = arow[g * 4 + 1]; }
        else { int ko = (g - KGX) * 4; a[0] = hrow[ko]; a[1] = hrow[ko + 1]; }
        const float2* wb = wf2 + (size_t)g * (48 * 32) + lane;
        #pragma unroll
        for (int i = 0; i < 8; ++i) {
          float2 bb = wb[(wid * 8 + i) * 32];
          v2f b; b[0] = bb.x; b[1] = bb.y;
          acc[i] = wmma4(a, b, acc[i]);
        }
      }
      #pragma unroll
      for (int i = 0; i < 8; ++i) {
        int nt = wid * 8 + i;
        #pragma unroll
        for (int v = 0; v < 8; ++v) Gh[(rbase + v) * GST + nt * 16 + ncol] = acc[i][v];
      }
    } else {
      v8f accx[4], acch[4];
      #pragma unroll
      for (int i = 0; i < 4; ++i) { accx[i] = v8zero(); acch[i] = v8zero(); }
      const int ntb = 32 + (wid - 4) * 4;
      for (int g = 0; g < KG; ++g) {
        v2f a;
        const bool xpart = (g < KGX);
        if (xpart) { a[0] = arow[g * 4]; a[1] = arow[g * 4 + 1]; }
        else { int ko = (g - KGX) * 4; a[0] = hrow[ko]; a[1] = hrow[ko + 1]; }
        const float2* wb = wf2 + (size_t)g * (48 * 32) + lane;
        #pragma unroll
        for (int i = 0; i < 4; ++i) {
          float2 bb = wb[(ntb + i) * 32];
          v2f b; b[0] = bb.x; b[1] = bb.y;
          if (xpart) accx[i] = wmma4(a, b, accx[i]);
          else       acch[i] = wmma4(a, b, acch[i]);
        }
      }
      #pragma unroll
      for (int i = 0; i < 4; ++i) {
        int nt = ntb + i;
        #pragma unroll
        for (int v = 0; v < 8; ++v) {
          Gh[(rbase + v) * GST + nt * 16 + ncol] = acch[i][v];
          Gin[(rbase + v) * HST + (nt - 32) * 16 + ncol] = accx[i][v];
        }
      }
    }
    __syncthreads();

    // gates + state update (each thread owns distinct (b,c) -> in-place Hs ok)
    for (int i = tid; i < 4096; i += 256) {
      int b = i >> 8, c = i & 255;
      float gr = Gh[b * GST + c] + bih[c] + bhh[c];
      float gz = Gh[b * GST + 256 + c] + bih[256 + c] + bhh[256 + c];
      float hn = Gh[b * GST + 512 + c] + bhh[512 + c];
      float xn = Gin[b * HST + c] + bih[512 + c];
      float r = sigm(gr);
      float z = sigm(gz);
      float nn = tanhf(xn + r * hn);
      float hN = (1.0f - z) * nn + z * Hs[b * HST + c];
      Hs[b * HST + c] = hN;
      if (yout) yout[(size_t)(s * 512 + n0 + b) * 512 + (size_t)dir * 256 + c] = hN;
    }
    __syncthreads();
  }

  if (hfinal) {
    for (int i = tid; i < 4096; i += 256) {
      int b = i >> 8, c = i & 255;
      hfinal[(size_t)(n0 + b) * 512 + (size_t)dir * 256 + c] = Hs[b * HST + c];
    }
  }
}

// ---------------------------------------------------------------------------
// dec_h = (combined[:64] @ w_adj.T + b_adj) : M=64, N=256, K=512, WMMA tiles
// ---------------------------------------------------------------------------
__global__ void __launch_bounds__(256)
adj_gemm(const float* __restrict__ A,    // combined (512,512); rows 0..63 used
         const float* __restrict__ W,    // w_adj (256,512)
         const float* __restrict__ bias, // (256)
         float* __restrict__ out) {      // (64,256)
  const int lane = threadIdx.x & 31;
  const int wid = blockIdx.x * 8 + (threadIdx.x >> 5); // 0..63
  const int mt = wid & 3;
  const int nt = wid >> 2; // 0..15
  const int mrow = mt * 16 + (lane & 15);
  const int ncol = nt * 16 + (lane & 15);
  const int koff = (lane >> 4) << 1;
  v8f acc = v8zero();
  for (int g = 0; g < 128; ++g) {
    int k = g * 4 + koff;
    v2f a; a[0] = A[(size_t)mrow * 512 + k]; a[1] = A[(size_t)mrow * 512 + k + 1];
    v2f b; b[0] = W[(size_t)ncol * 512 + k]; b[1] = W[(size_t)ncol * 512 + k + 1];
    acc = wmma4(a, b, acc);
  }
  #pragma unroll
  for (int v = 0; v < 8; ++v) {
    int m = mt * 16 + v + ((lane >> 4) << 3);
    int n = nt * 16 + (lane & 15);
    out[(size_t)m * 256 + n] = acc[v] + bias[n];
  }
}

// ---------------------------------------------------------------------------
// Decoder GRU step: batch 64, H=256, input 56 (null => zeros at step 0)
// ---------------------------------------------------------------------------
__global__ void __launch_bounds__(256)
dec_step(const float* __restrict__ inp, const float* __restrict__ hin,
         const float* __restrict__ wih,  // (768,56)
         const float* __restrict__ whh,  // (768,256)
         const float* __restrict__ bih, const float* __restrict__ bhh,
         float* __restrict__ hout) {
  __shared__ float hrow[256];
  __shared__ float irow[56];
  const int b = blockIdx.x;
  const int j = threadIdx.x;
  hrow[j] = hin[b * 256 + j];
  if (j < 56) irow[j] = inp ? inp[b * 56 + j] : 0.0f;
  __syncthreads();
  float gi_r = bih[j], gi_z = bih[256 + j], gi_n = bih[512 + j];
  if (inp) {
    for (int k = 0; k < 56; ++k) {
      float xv = irow[k];
      gi_r += xv * wih[j * 56 + k];
      gi_z += xv * wih[(256 + j) * 56 + k];
      gi_n += xv * wih[(512 + j) * 56 + k];
    }
  }
  float gh_r = bhh[j], gh_z = bhh[256 + j], gh_n = bhh[512 + j];
  for (int k = 0; k < 256; ++k) {
    float hv = hrow[k];
    gh_r += hv * whh[j * 256 + k];
    gh_z += hv * whh[(256 + j) * 256 + k];
    gh_n += hv * whh[(512 + j) * 256 + k];
  }
  float r = sigm(gi_r + gh_r);
  float z = sigm(gi_z + gh_z);
  float n = tanhf(gi_n + r * gh_n);
  hout[b * 256 + j] = (1.0f - z) * n + z * hrow[j];
}

__global__ void __launch_bounds__(64)
dec_out(const float* __restrict__ h, const float* __restrict__ wfc, // (56,256)
        const float* __restrict__ bfc, float* __restrict__ inpbuf,
        float* __restrict__ dout, int step) {
  __shared__ float hrow[256];
  const int b = blockIdx.x;
  const int t = threadIdx.x;
  for (int i = t; i < 256; i += 64) hrow[i] = h[b * 256 + i];
  __syncthreads();
  if (t < 56) {
    float acc = bfc[t];
    for (int k = 0; k < 256; ++k) acc += hrow[k] * wfc[t * 256 + k];
    inpbuf[b * 56 + t] = acc;
    dout[b * 336 + step * 56 + t] = acc; // output layout (64, 6, 56)
  }
}

// ---------------------------------------------------------------------------
extern "C" void kernel_launch(void* const* d_in, const int* in_sizes, int n_in,
                              void* d_out, int out_size, void* d_ws, size_t ws_size,
                              hipStream_t stream) {
  (void)in_sizes; (void)n_in; (void)out_size; (void)ws_size;
  const float* x       = (const float*)d_in[0];
  const float* w_ih_1f = (const float*)d_in[1];
  const float* w_hh_1f = (const float*)d_in[2];
  const float* b_ih_1f = (const float*)d_in[3];
  const float* b_hh_1f = (const float*)d_in[4];
  const float* w_ih_1b = (const float*)d_in[5];
  const float* w_hh_1b = (const float*)d_in[6];
  const float* b_ih_1b = (const float*)d_in[7];
  const float* b_hh_1b = (const float*)d_in[8];
  const float* w_ih_2f = (const float*)d_in[9];
  const float* w_hh_2f = (const float*)d_in[10];
  const float* b_ih_2f = (const float*)d_in[11];
  const float* b_hh_2f = (const float*)d_in[12];
  const float* w_ih_2b = (const float*)d_in[13];
  const float* w_hh_2b = (const float*)d_in[14];
  const float* b_ih_2b = (const float*)d_in[15];
  const float* b_hh_2b = (const float*)d_in[16];
  const float* w_ih_d  = (const float*)d_in[17];
  const float* w_hh_d  = (const float*)d_in[18];
  const float* b_ih_d  = (const float*)d_in[19];
  const float* b_hh_d  = (const float*)d_in[20];
  const float* w_fc1   = (const float*)d_in[21];
  const float* b_fc1   = (const float*)d_in[22];
  const float* w_adj   = (const float*)d_in[23];
  const float* b_adj   = (const float*)d_in[24];
  float* out = (float*)d_out;

  float* ws = (float*)d_ws;
  size_t off = 0;
  float* seq  = ws + off; off += (size_t)32768 * 192;  // 6.29 M
  float* y1   = ws + off; off += (size_t)32768 * 512;  // 16.78 M
  float* comb = ws + off; off += (size_t)512 * 512;
  float* wf1f = ws + off; off += (size_t)448 * 768;
  float* wf1b = ws + off; off += (size_t)448 * 768;
  float* wf2f = ws + off; off += (size_t)768 * 768;
  float* wf2b = ws + off; off += (size_t)768 * 768;
  float* hA   = ws + off; off += (size_t)64 * 256;
  float* hB   = ws + off; off += (size_t)64 * 256;
  float* inpb = ws + off; off += (size_t)64 * 56;

  make_seq<<<2048, 256, 0, stream>>>(x, seq);
  reorder_w<<<512, 256, 0, stream>>>(w_ih_1f, w_hh_1f, wf1f, 192);
  reorder_w<<<512, 256, 0, stream>>>(w_ih_1b, w_hh_1b, wf1b, 192);
  reorder_w<<<512, 256, 0, stream>>>(w_ih_2f, w_hh_2f, wf2f, 512);
  reorder_w<<<512, 256, 0, stream>>>(w_ih_2b, w_hh_2b, wf2b, 512);

  gru_layer_kernel<192><<<dim3(32, 2), 256, 0, stream>>>(
      seq, wf1f, wf1b, b_ih_1f, b_hh_1f, b_ih_1b, b_hh_1b, y1, nullptr);
  gru_layer_kernel<512><<<dim3(32, 2), 256, 0, stream>>>(
      y1, wf2f, wf2b, b_ih_2f, b_hh_2f, b_ih_2b, b_hh_2b, nullptr, comb);

  adj_gemm<<<8, 256, 0, stream>>>(comb, w_adj, b_adj, hA);

  for (int step = 0; step < 6; ++step) {
    const float* inp = step ? inpb : nullptr;
    const float* hin = (step & 1) ? hB : hA;
    float* hout = (step & 1) ? hA : hB;
    dec_step<<<64, 256, 0, stream>>>(inp, hin, w_ih_d, w_hh_d, b_ih_d, b_hh_d, hout);
    dec_out<<<64, 64, 0, stream>>>(hout, w_fc1, b_fc1, inpb, out, step);
  }
}